// KBpA_89232240542412
// MI455X (gfx1250) — compile-verified
//
#include <hip/hip_runtime.h>
#include <hip/hip_bf16.h>

// ---------------------------------------------------------------------------
// Problem constants (from reference)
// ---------------------------------------------------------------------------
#define N_PIX 65536
#define N_CEN 2048
#define N_G   1024
// exp(-d2/(2*SDP2)) = exp2(k * d2),  k = -1/(2*SDP2*ln2), SDP2 = 1.0
#define KCONST (-0.7213475204444817f)

typedef __attribute__((ext_vector_type(2))) float v2f;
typedef __attribute__((ext_vector_type(8))) float v8f;

// ---------------------------------------------------------------------------
// Phase 1: deformation = K_def @ betas ; dp = all_pixels - deformation
// Memory-bound (256 MB stream of K_def). One wave32 per pixel row, coalesced
// float4 (global_load_b128) reads, betas cached in LDS, shfl_xor reduction.
// Writes dp[N_PIX][2] into workspace.
// ---------------------------------------------------------------------------
__global__ __launch_bounds__(256) void deform_kernel(
    const float* __restrict__ betas,       // [N_G,2]
    const float* __restrict__ K_def,       // [N_PIX,N_G]
    const float* __restrict__ all_pixels,  // [N_PIX,2]
    float* __restrict__ dp_out)            // [N_PIX,2] (workspace)
{
    __shared__ float2 shb[N_G];            // betas as float2 per g-center (8KB)

    const int tid = threadIdx.x;
    const float2* b2 = (const float2*)betas;
    #pragma unroll
    for (int k = 0; k < N_G / 256; ++k)
        shb[tid + k * 256] = b2[tid + k * 256];
    __syncthreads();

    const int lane = tid & 31;
    const int wave = tid >> 5;
    const int row  = blockIdx.x * 8 + wave;   // 8192 blocks * 8 waves = 65536

    const float4* kd4 = (const float4*)(K_def + (size_t)row * N_G);

    float acc0 = 0.0f, acc1 = 0.0f;
    #pragma unroll
    for (int i = 0; i < N_G / 128; ++i) {     // 8 iterations
        const int q  = i * 32 + lane;         // float4 index within row
        const float4 kv = kd4[q];             // coalesced b128
        const int c  = q * 4;
        const float2 g0 = shb[c + 0];
        const float2 g1 = shb[c + 1];
        const float2 g2 = shb[c + 2];
        const float2 g3 = shb[c + 3];
        acc0 = fmaf(kv.x, g0.x, acc0); acc1 = fmaf(kv.x, g0.y, acc1);
        acc0 = fmaf(kv.y, g1.x, acc0); acc1 = fmaf(kv.y, g1.y, acc1);
        acc0 = fmaf(kv.z, g2.x, acc0); acc1 = fmaf(kv.z, g2.y, acc1);
        acc0 = fmaf(kv.w, g3.x, acc0); acc1 = fmaf(kv.w, g3.y, acc1);
    }

    // wave32 all-reduce
    #pragma unroll
    for (int m = 16; m >= 1; m >>= 1) {
        acc0 += __shfl_xor(acc0, m, 32);
        acc1 += __shfl_xor(acc1, m, 32);
    }

    if (lane == 0) {
        const float px = all_pixels[row * 2 + 0];
        const float py = all_pixels[row * 2 + 1];
        dp_out[row * 2 + 0] = px - acc0;
        dp_out[row * 2 + 1] = py - acc1;
    }
}

// ---------------------------------------------------------------------------
// Phase 2: out[n] = sum_m exp2(k*d2(n,m)) * alpha[m]
// d2 computed as a rank-4 contraction per 16x16 tile via V_WMMA_F32_16X16X4_F32:
//   A row (pixel)  = [dpx, dpy, |dp|^2, 1]
//   B col (center) = k * [-2cx, -2cy, 1, |c|^2]
// so D = k*d2 is the exp2 exponent directly.
//
// VGPR layout (ISA 7.12.2, 32-bit 16x4 operands):
//   VGPR0: lanes 0-15 -> K=0, lanes 16-31 -> K=2
//   VGPR1: lanes 0-15 -> K=1, lanes 16-31 -> K=3
// D (16x16 f32): lane = column (center), VGPR r = row (pixel r, +8 for hi lanes)
//
// One wave per 16-pixel tile; 2048 centers preprocessed into 32KB LDS.
// ---------------------------------------------------------------------------
__global__ __launch_bounds__(256) void rbf_kernel(
    const float* __restrict__ dp,        // [N_PIX,2] from workspace
    const float* __restrict__ centers,   // [N_CEN,2]
    const float* __restrict__ alphas,    // [N_CEN]
    float* __restrict__ out)             // [N_PIX]
{
    __shared__ float4 shc[N_CEN];        // {-2k*cx, -2k*cy, k*|c|^2, alpha} 32KB

    const int tid = threadIdx.x;
    #pragma unroll
    for (int k = 0; k < N_CEN / 256; ++k) {
        const int j = tid + k * 256;
        const float cx = centers[j * 2 + 0];
        const float cy = centers[j * 2 + 1];
        shc[j] = make_float4(-2.0f * KCONST * cx,
                             -2.0f * KCONST * cy,
                             KCONST * (cx * cx + cy * cy),
                             alphas[j]);
    }
    __syncthreads();

    const int lane = tid & 31;
    const int wave = tid >> 5;
    const int pt   = blockIdx.x * 8 + wave;   // 512 blocks * 8 waves = 4096 tiles
    const int lo16 = lane & 15;
    const bool isLow = lane < 16;

    // A fragment: pixel p = pt*16 + (lane&15)
    const int p = pt * 16 + lo16;
    const float dpx = dp[p * 2 + 0];
    const float dpy = dp[p * 2 + 1];
    v2f a;
    a.x = isLow ? dpx : fmaf(dpx, dpx, dpy * dpy);  // K0 : K2 (=|dp|^2)
    a.y = isLow ? dpy : 1.0f;                       // K1 : K3 (=1)

    float acc[8];
    #pragma unroll
    for (int r = 0; r < 8; ++r) acc[r] = 0.0f;

    const v8f czero = {};

    for (int t = 0; t < N_CEN / 16; ++t) {          // 128 center tiles
        const float4 b4 = shc[t * 16 + lo16];       // ds_load_b128
        v2f b;
        b.x = isLow ? b4.x : KCONST;                // K0 : K2 (=k)
        b.y = isLow ? b4.y : b4.z;                  // K1 : K3 (=k*|c|^2)

        // D[r] = k * d2(pixel r + 8*(lane/16), center t*16 + lane%16)
        v8f d = __builtin_amdgcn_wmma_f32_16x16x4_f32(
            false, a, false, b, (short)0, czero, false, false);

        const float alpha = b4.w;                   // same center col for hi/lo lanes
        #pragma unroll
        for (int r = 0; r < 8; ++r)
            acc[r] = fmaf(__builtin_amdgcn_exp2f(d[r]), alpha, acc[r]);
    }

    // Reduce across the 16 center-columns (within each 16-lane half)
    #pragma unroll
    for (int r = 0; r < 8; ++r) {
        float v = acc[r];
        v += __shfl_xor(v, 1, 16);
        v += __shfl_xor(v, 2, 16);
        v += __shfl_xor(v, 4, 16);
        v += __shfl_xor(v, 8, 16);
        acc[r] = v;
    }

    if (lo16 == 0) {
        const int base = pt * 16 + ((lane >> 4) << 3);  // lane0 -> rows 0..7, lane16 -> 8..15
        #pragma unroll
        for (int r = 0; r < 8; ++r)
            out[base + r] = acc[r];
    }
}

// ---------------------------------------------------------------------------
extern "C" void kernel_launch(void* const* d_in, const int* in_sizes, int n_in,
                              void* d_out, int out_size, void* d_ws, size_t ws_size,
                              hipStream_t stream) {
    const float* betas       = (const float*)d_in[0];   // [1024,2]
    const float* K_def       = (const float*)d_in[1];   // [65536,1024]
    const float* all_pixels  = (const float*)d_in[2];   // [65536,2]
    const float* all_centers = (const float*)d_in[3];   // [2048,2]
    const float* alphas      = (const float*)d_in[4];   // [2048,1]
    float* out = (float*)d_out;                          // [65536,1]
    float* dp  = (float*)d_ws;                           // [65536,2] scratch

    deform_kernel<<<dim3(N_PIX / 8), dim3(256), 0, stream>>>(
        betas, K_def, all_pixels, dp);

    rbf_kernel<<<dim3(N_PIX / 16 / 8), dim3(256), 0, stream>>>(
        dp, all_centers, alphas, out);
}